// GCN_67396626809328
// MI455X (gfx1250) — compile-verified
//
#include <hip/hip_runtime.h>
#include <hip/hip_bf16.h>

typedef __attribute__((ext_vector_type(2))) float v2f;
typedef __attribute__((ext_vector_type(8))) float v8f;

#define NNODES 50000
#define NEDGES 800000
#define HDIM   128
#define NGRAPH 256
#define NOUT   10

// ---------------- degree / norm ----------------
__global__ void deg_init(float* deg, int n) {
    int i = blockIdx.x * blockDim.x + threadIdx.x;
    if (i < n) deg[i] = 1.0f;           // self-loop contributes 1
}
__global__ void deg_edges(float* deg, const int* __restrict__ dst, int e) {
    int i = blockIdx.x * blockDim.x + threadIdx.x;
    if (i < e) atomicAdd(&deg[dst[i]], 1.0f);
}
__global__ void deg_rsqrt(float* deg, int n) {
    int i = blockIdx.x * blockDim.x + threadIdx.x;
    if (i < n) deg[i] = rsqrtf(deg[i]);     // dis = D^-1/2
}

// ---------------- GEMM: C[n,128] = A[n,128] @ W[128,128]  (f32 WMMA) ----------------
// One wave computes a 16x128 strip. A fragment reused across 8 column tiles.
__global__ void gemm_f32_wmma(const float* __restrict__ A,
                              const float* __restrict__ W,
                              float* __restrict__ C, int nrows) {
    int wave = (blockIdx.x * blockDim.x + threadIdx.x) >> 5;
    int lane = threadIdx.x & 31;
    int row0 = wave * 16;
    if (row0 >= nrows) return;          // wave-uniform: EXEC stays all-ones
    int half = lane >> 4;               // 0: lanes 0-15, 1: lanes 16-31
    int li   = lane & 15;

    v8f acc[8];
#pragma unroll
    for (int t = 0; t < 8; ++t) acc[t] = (v8f){0,0,0,0,0,0,0,0};

    const float* arow = A + (size_t)(row0 + li) * HDIM;
    for (int k0 = 0; k0 < HDIM; k0 += 4) {
        // A 16x4 fragment: lane<16 holds K=k0+0/1, lane>=16 holds K=k0+2/3
        v2f a;
        a.x = arow[k0 + 2 * half + 0];
        a.y = arow[k0 + 2 * half + 1];
#pragma unroll
        for (int t = 0; t < 8; ++t) {
            int nb = t * 16;
            // B 4x16 fragment: VGPR0 = rows K=k0+2h, VGPR1 = rows K=k0+2h+1
            v2f b;
            b.x = W[(size_t)(k0 + 2 * half + 0) * HDIM + nb + li];
            b.y = W[(size_t)(k0 + 2 * half + 1) * HDIM + nb + li];
            acc[t] = __builtin_amdgcn_wmma_f32_16x16x4_f32(
                false, a, false, b, (short)0, acc[t], false, false);
        }
    }
    // C/D layout: reg r -> row (r + 8*half), col li within tile
#pragma unroll
    for (int t = 0; t < 8; ++t) {
#pragma unroll
        for (int r = 0; r < 8; ++r) {
            C[(size_t)(row0 + r + 8 * half) * HDIM + t * 16 + li] = acc[t][r];
        }
    }
}

// ---------------- self-loop + bias init of aggregate ----------------
__global__ void selfloop_bias(const float* __restrict__ h, float* __restrict__ agg,
                              const float* __restrict__ dis, const float* __restrict__ b,
                              int total) {
    int i = blockIdx.x * blockDim.x + threadIdx.x;
    if (i >= total) return;
    int node = i >> 7;
    int f    = i & 127;
    float di = dis[node];
    agg[i] = h[i] * di * di + b[f];
}

// ---------------- edge scatter: agg[dst] += norm * h[src]  (wave per edge) ----------------
__global__ void scatter_edges(const float* __restrict__ h, float* __restrict__ agg,
                              const int* __restrict__ src, const int* __restrict__ dst,
                              const float* __restrict__ dis, int nedges) {
    int wave = (blockIdx.x * blockDim.x + threadIdx.x) >> 5;
    int lane = threadIdx.x & 31;
    if (wave >= nedges) return;
    int s = src[wave];
    int d = dst[wave];
    float nrm = dis[s] * dis[d];
    const float4* hp = (const float4*)(h + (size_t)s * HDIM);
    float4 v = hp[lane];
    float* ap = agg + (size_t)d * HDIM + lane * 4;
    atomicAdd(ap + 0, v.x * nrm);
    atomicAdd(ap + 1, v.y * nrm);
    atomicAdd(ap + 2, v.z * nrm);
    atomicAdd(ap + 3, v.w * nrm);
}

// ---------------- batchnorm ----------------
__global__ void zero_f32(float* p, int n) {
    int i = blockIdx.x * blockDim.x + threadIdx.x;
    if (i < n) p[i] = 0.0f;
}
__global__ void bn_stats(const float* __restrict__ x, float* sum, float* sumsq, int n) {
    int f = threadIdx.x;                    // 128 threads = 128 features
    float s = 0.f, sq = 0.f;
    for (int r = blockIdx.x; r < n; r += gridDim.x) {
        float v = x[(size_t)r * HDIM + f];
        s  += v;
        sq += v * v;
    }
    atomicAdd(&sum[f],   s);
    atomicAdd(&sumsq[f], sq);
}
__global__ void bn_finalize(const float* sum, const float* sumsq,
                            const float* __restrict__ g, const float* __restrict__ be,
                            float* scale, float* shift, float n) {
    int f = threadIdx.x;
    float mu  = sum[f] / n;
    float var = sumsq[f] / n - mu * mu;
    float inv = rsqrtf(var + 1e-5f);
    float sc  = inv * g[f];
    scale[f] = sc;
    shift[f] = be[f] - mu * sc;
}
__global__ void bn_relu(const float* __restrict__ x, float* __restrict__ y,
                        const float* __restrict__ scale, const float* __restrict__ shift,
                        int total) {
    int i = blockIdx.x * blockDim.x + threadIdx.x;
    if (i >= total) return;
    int f = i & 127;
    float v = x[i] * scale[f] + shift[f];
    y[i] = v > 0.f ? v : 0.f;
}

// ---------------- global mean pool (wave per node) ----------------
__global__ void pool_nodes(const float* __restrict__ h, const int* __restrict__ batch,
                           float* pooled, float* counts, int n) {
    int wave = (blockIdx.x * blockDim.x + threadIdx.x) >> 5;
    int lane = threadIdx.x & 31;
    if (wave >= n) return;
    int g = batch[wave];
    const float4* hp = (const float4*)(h + (size_t)wave * HDIM);
    float4 v = hp[lane];
    float* pp = pooled + (size_t)g * HDIM + lane * 4;
    atomicAdd(pp + 0, v.x);
    atomicAdd(pp + 1, v.y);
    atomicAdd(pp + 2, v.z);
    atomicAdd(pp + 3, v.w);
    if (lane == 0) atomicAdd(&counts[g], 1.0f);
}

// ---------------- output head: out = (pooled/cnt) @ Wout + bout ----------------
__global__ void out_head(const float* __restrict__ pooled, const float* __restrict__ counts,
                         const float* __restrict__ Wout, const float* __restrict__ bout,
                         float* __restrict__ out) {
    int i = blockIdx.x * blockDim.x + threadIdx.x;
    if (i >= NGRAPH * NOUT) return;
    int g = i / NOUT;
    int o = i % NOUT;
    float c = counts[g];
    c = c > 1.0f ? c : 1.0f;
    float acc = 0.f;
#pragma unroll 8
    for (int k = 0; k < HDIM; ++k)
        acc += pooled[(size_t)g * HDIM + k] * Wout[k * NOUT + o];
    out[i] = acc / c + bout[o];
}

extern "C" void kernel_launch(void* const* d_in, const int* in_sizes, int n_in,
                              void* d_out, int out_size, void* d_ws, size_t ws_size,
                              hipStream_t stream) {
    const float* x    = (const float*)d_in[0];
    const int*   ei   = (const int*)  d_in[1];   // [2,E]: src then dst
    const int*   batch= (const int*)  d_in[2];
    const float* W1   = (const float*)d_in[3];
    const float* b1   = (const float*)d_in[4];
    const float* g1   = (const float*)d_in[5];
    const float* be1  = (const float*)d_in[6];
    const float* W2   = (const float*)d_in[7];
    const float* b2   = (const float*)d_in[8];
    const float* g2   = (const float*)d_in[9];
    const float* be2  = (const float*)d_in[10];
    const float* Wout = (const float*)d_in[11];
    const float* bout = (const float*)d_in[12];
    float* out = (float*)d_out;

    const int* src = ei;
    const int* dst = ei + NEDGES;

    // workspace carve-up
    float* ws    = (float*)d_ws;
    float* bufA  = ws;                               // N*128
    float* bufB  = bufA + (size_t)NNODES * HDIM;     // N*128
    float* dis   = bufB + (size_t)NNODES * HDIM;     // N
    float* sum   = dis + NNODES;                     // 128
    float* sumsq = sum + HDIM;                       // 128
    float* scale = sumsq + HDIM;                     // 128
    float* shift = scale + HDIM;                     // 128
    float* pooled= shift + HDIM;                     // G*128
    float* counts= pooled + (size_t)NGRAPH * HDIM;   // G

    const int TOT = NNODES * HDIM;
    const int B256 = 256;
    int gemm_threads = (NNODES / 16) * 32;           // 3125 waves
    int gemm_blocks  = (gemm_threads + B256 - 1) / B256;
    int scat_blocks  = (NEDGES * 32 + B256 - 1) / B256;
    int elem_blocks  = (TOT + B256 - 1) / B256;
    int pool_blocks  = (NNODES * 32 + B256 - 1) / B256;

    // degrees -> dis = rsqrt(deg)
    deg_init <<<(NNODES + B256 - 1) / B256, B256, 0, stream>>>(dis, NNODES);
    deg_edges<<<(NEDGES + B256 - 1) / B256, B256, 0, stream>>>(dis, dst, NEDGES);
    deg_rsqrt<<<(NNODES + B256 - 1) / B256, B256, 0, stream>>>(dis, NNODES);

    // ---- layer 1 ----
    gemm_f32_wmma<<<gemm_blocks, B256, 0, stream>>>(x, W1, bufA, NNODES);
    selfloop_bias<<<elem_blocks, B256, 0, stream>>>(bufA, bufB, dis, b1, TOT);
    scatter_edges<<<scat_blocks, B256, 0, stream>>>(bufA, bufB, src, dst, dis, NEDGES);
    zero_f32<<<1, 256, 0, stream>>>(sum, 2 * HDIM);          // sum+sumsq contiguous
    bn_stats<<<256, HDIM, 0, stream>>>(bufB, sum, sumsq, NNODES);
    bn_finalize<<<1, HDIM, 0, stream>>>(sum, sumsq, g1, be1, scale, shift, (float)NNODES);
    bn_relu<<<elem_blocks, B256, 0, stream>>>(bufB, bufA, scale, shift, TOT);

    // ---- layer 2 ----
    gemm_f32_wmma<<<gemm_blocks, B256, 0, stream>>>(bufA, W2, bufB, NNODES);
    selfloop_bias<<<elem_blocks, B256, 0, stream>>>(bufB, bufA, dis, b2, TOT);
    scatter_edges<<<scat_blocks, B256, 0, stream>>>(bufB, bufA, src, dst, dis, NEDGES);
    zero_f32<<<1, 256, 0, stream>>>(sum, 2 * HDIM);
    bn_stats<<<256, HDIM, 0, stream>>>(bufA, sum, sumsq, NNODES);
    bn_finalize<<<1, HDIM, 0, stream>>>(sum, sumsq, g2, be2, scale, shift, (float)NNODES);
    bn_relu<<<elem_blocks, B256, 0, stream>>>(bufA, bufB, scale, shift, TOT);

    // ---- pool + head ----
    zero_f32<<<(NGRAPH * HDIM + NGRAPH + B256 - 1) / B256, B256, 0, stream>>>(pooled, NGRAPH * HDIM + NGRAPH);
    pool_nodes<<<pool_blocks, B256, 0, stream>>>(bufB, batch, pooled, counts, NNODES);
    out_head<<<(NGRAPH * NOUT + B256 - 1) / B256, B256, 0, stream>>>(pooled, counts, Wout, bout, out);
}